// PScan_82291573391742
// MI455X (gfx1250) — compile-verified
//
#include <hip/hip_runtime.h>

#ifndef __has_builtin
#define __has_builtin(x) 0
#endif

// Problem shape (fixed by the reference): B=4, L=2048, C=64, D=4.
constexpr int kB = 4;
constexpr int kL = 2048;
constexpr int kC = 64;
constexpr int kD = 4;

constexpr int TPB        = 256;               // 8 wave32 per workgroup
constexpr int WIN        = kL / TPB;          // 8 timesteps per thread
constexpr int SLOT_BYTES = WIN * 16 + 16;     // 8 x float4 + 16B pad (bank de-conflict)
constexpr int XRE_OFF    = 0;
constexpr int XIM_OFF    = TPB * SLOT_BYTES;          // 36864
constexpr int SCAN_OFF   = 2 * TPB * SLOT_BYTES;      // 73728
constexpr int AGG_BYTES  = 48;                        // 10 floats + pad, 16B aligned
constexpr int LDS_BYTES  = SCAN_OFF + TPB * AGG_BYTES; // 86016 (needs CDNA5's big LDS)

typedef float v4f __attribute__((ext_vector_type(4)));
typedef int   v4i __attribute__((ext_vector_type(4)));
typedef __attribute__((address_space(1))) v4i gv4i;   // gv4i* == int4 addrspace(1)*
typedef __attribute__((address_space(3))) v4i lv4i;   // lv4i* == int4 addrspace(3)*
typedef __attribute__((address_space(3))) char lchar;
typedef __attribute__((address_space(3))) v4f  lv4f_t;

// ---- CDNA5 async global->LDS gather (ASYNCcnt path) -------------------------
__device__ __forceinline__ void async_gather_b128(const float* gsrc, lchar* ldst) {
#if __has_builtin(__builtin_amdgcn_global_load_async_to_lds_b128)
  __builtin_amdgcn_global_load_async_to_lds_b128((gv4i*)gsrc, (lv4i*)ldst,
                                                 /*offset=*/0, /*cpol=*/0);
#else
  // Fallback: raw gfx1250 encoding. VDST = LDS byte address, VADDR = 64-bit vaddr.
  asm volatile("global_load_async_to_lds_b128 %0, %1, off"
               :: "v"(ldst), "v"(gsrc) : "memory");
#endif
}

__device__ __forceinline__ void wait_async0() {
#if __has_builtin(__builtin_amdgcn_s_wait_asynccnt)
  __builtin_amdgcn_s_wait_asynccnt(0);
#else
  asm volatile("s_wait_asynccnt 0" ::: "memory");
#endif
}

// ---- Kernel: one workgroup per (b, c, d) lane -------------------------------
__global__ __launch_bounds__(TPB) void pscan_diag_complex(
    const float* __restrict__ A_re, const float* __restrict__ A_im,
    const float* __restrict__ X_re, const float* __restrict__ X_im,
    float* __restrict__ Y) {
  extern __shared__ char smem_raw[];
  lchar* smem = (lchar*)smem_raw;

  const int tid = threadIdx.x;
  const int g   = blockIdx.x;     // b*(C*D) + c*D + d
  const int b   = g >> 8;         // / 256
  const int cd  = g & 255;        // c*D + d

  const int t0 = tid * WIN;

  const int xstride = kC * kD * kD;                  // 1024 floats per timestep
  const int xbase   = b * (kL * xstride) + cd * kD;  // e = 0
  const int astride = kC * kD;                       // 256 floats per timestep
  const int abase   = b * (kL * astride) + cd;

  const int slot = tid * SLOT_BYTES;

  // Stage 1: async-gather this thread's X window into LDS (16 outstanding b128)
#pragma unroll
  for (int i = 0; i < WIN; ++i) {
    const int t = t0 + i;
    async_gather_b128(X_re + xbase + t * xstride, smem + (XRE_OFF + slot + i * 16));
    async_gather_b128(X_im + xbase + t * xstride, smem + (XIM_OFF + slot + i * 16));
  }

  // Stage 2: A window -> registers (overlaps with the async copies)
  float ar[WIN], ai[WIN];
#pragma unroll
  for (int i = 0; i < WIN; ++i) {
    const int t = t0 + i;
    ar[i] = A_re[abase + t * astride];
    ai[i] = A_im[abase + t * astride];
  }

  wait_async0();  // this wave's LDS tiles are ready; each lane reads only its own

  // Stage 3: thread-local window aggregate (state_in = 0):
  //   transform of the window is  s -> pa*s + y
  float pa_r = 1.f, pa_i = 0.f;
  float yr[kD] = {0.f, 0.f, 0.f, 0.f}, yi[kD] = {0.f, 0.f, 0.f, 0.f};
#pragma unroll
  for (int i = 0; i < WIN; ++i) {
    v4f xr = *(lv4f_t*)(smem + (XRE_OFF + slot + i * 16));
    v4f xi = *(lv4f_t*)(smem + (XIM_OFF + slot + i * 16));
    const float a_r = ar[i], a_i = ai[i];
#pragma unroll
    for (int e = 0; e < kD; ++e) {
      const float nr = a_r * yr[e] - a_i * yi[e] + xr[e];
      const float ni = a_r * yi[e] + a_i * yr[e] + xi[e];
      yr[e] = nr; yi[e] = ni;
    }
    const float npr = a_r * pa_r - a_i * pa_i;
    const float npi = a_r * pa_i + a_i * pa_r;
    pa_r = npr; pa_i = npi;
  }

  // Stage 4: Hillis-Steele scan of 256 aggregates in LDS.
  // combine(prev, cur): a = a_c*a_p ; y[e] = a_c*y_p[e] + y_c[e]   (complex)
  lchar* scan = smem + SCAN_OFF;
  {
    lchar* mine = scan + tid * AGG_BYTES;
    v4f w0 = {pa_r, pa_i, yr[0], yr[1]};
    v4f w1 = {yr[2], yr[3], yi[0], yi[1]};
    v4f w2 = {yi[2], yi[3], 0.f, 0.f};
    *(lv4f_t*)(mine) = w0; *(lv4f_t*)(mine + 16) = w1; *(lv4f_t*)(mine + 32) = w2;
  }
  __syncthreads();
#pragma unroll
  for (int ofs = 1; ofs < TPB; ofs <<= 1) {
    float oar = 0.f, oai = 0.f, oyr[kD], oyi[kD];
    const bool has = (tid >= ofs);
    if (has) {
      lchar* p = scan + (tid - ofs) * AGG_BYTES;
      v4f q0 = *(lv4f_t*)(p);
      v4f q1 = *(lv4f_t*)(p + 16);
      v4f q2 = *(lv4f_t*)(p + 32);
      oar = q0[0]; oai = q0[1];
      oyr[0] = q0[2]; oyr[1] = q0[3]; oyr[2] = q1[0]; oyr[3] = q1[1];
      oyi[0] = q1[2]; oyi[1] = q1[3]; oyi[2] = q2[0]; oyi[3] = q2[1];
    }
    __syncthreads();
    if (has) {
#pragma unroll
      for (int e = 0; e < kD; ++e) {
        const float nr = pa_r * oyr[e] - pa_i * oyi[e] + yr[e];
        const float ni = pa_r * oyi[e] + pa_i * oyr[e] + yi[e];
        yr[e] = nr; yi[e] = ni;
      }
      const float nar = pa_r * oar - pa_i * oai;
      const float nai = pa_r * oai + pa_i * oar;
      pa_r = nar; pa_i = nai;
      lchar* mine = scan + tid * AGG_BYTES;
      v4f w0 = {pa_r, pa_i, yr[0], yr[1]};
      v4f w1 = {yr[2], yr[3], yi[0], yi[1]};
      v4f w2 = {yi[2], yi[3], 0.f, 0.f};
      *(lv4f_t*)(mine) = w0; *(lv4f_t*)(mine + 16) = w1; *(lv4f_t*)(mine + 32) = w2;
    }
    __syncthreads();
  }

  // Exclusive prefix = inclusive[tid-1] (identity for tid 0); only y is needed.
  float sr[kD] = {0.f, 0.f, 0.f, 0.f}, si[kD] = {0.f, 0.f, 0.f, 0.f};
  if (tid > 0) {
    lchar* p = scan + (tid - 1) * AGG_BYTES;
    v4f q0 = *(lv4f_t*)(p);
    v4f q1 = *(lv4f_t*)(p + 16);
    v4f q2 = *(lv4f_t*)(p + 32);
    sr[0] = q0[2]; sr[1] = q0[3]; sr[2] = q1[0]; sr[3] = q1[1];
    si[0] = q1[2]; si[1] = q1[3]; si[2] = q2[0]; si[3] = q2[1];
  }

  // Stage 5: replay the window seeded with the prefix; stream out [.., D, 2]
  const int obase = xbase * 2;
#pragma unroll
  for (int i = 0; i < WIN; ++i) {
    const int t = t0 + i;
    v4f xr = *(lv4f_t*)(smem + (XRE_OFF + slot + i * 16));
    v4f xi = *(lv4f_t*)(smem + (XIM_OFF + slot + i * 16));
    const float a_r = ar[i], a_i = ai[i];
#pragma unroll
    for (int e = 0; e < kD; ++e) {
      const float nr = a_r * sr[e] - a_i * si[e] + xr[e];
      const float ni = a_r * si[e] + a_i * sr[e] + xi[e];
      sr[e] = nr; si[e] = ni;
    }
    float* o = Y + obase + t * (xstride * 2);
    v4f o0 = {sr[0], si[0], sr[1], si[1]};
    v4f o1 = {sr[2], si[2], sr[3], si[3]};
    __builtin_nontemporal_store(o0, (v4f*)o);        // streamed, never re-read
    __builtin_nontemporal_store(o1, (v4f*)(o + 4));
  }
}

extern "C" void kernel_launch(void* const* d_in, const int* in_sizes, int n_in,
                              void* d_out, int out_size, void* d_ws, size_t ws_size,
                              hipStream_t stream) {
  (void)in_sizes; (void)n_in; (void)out_size; (void)d_ws; (void)ws_size;
  const float* A_re = (const float*)d_in[0];
  const float* A_im = (const float*)d_in[1];
  const float* X_re = (const float*)d_in[2];
  const float* X_im = (const float*)d_in[3];
  float* Y = (float*)d_out;

  // 86 KB dynamic LDS per workgroup: legal on CDNA5 (320 KB/WGP), raise the cap.
  (void)hipFuncSetAttribute((const void*)pscan_diag_complex,
                            hipFuncAttributeMaxDynamicSharedMemorySize, LDS_BYTES);

  dim3 grid(kB * kC * kD);   // 1024 workgroups: one per (b, c, d) lane
  dim3 block(TPB);
  pscan_diag_complex<<<grid, block, LDS_BYTES, stream>>>(A_re, A_im, X_re, X_im, Y);
}